// CombinedModel_12953621365421
// MI455X (gfx1250) — compile-verified
//
#include <hip/hip_runtime.h>
#include <hip/hip_bf16.h>

// out[n,:] = relu( x[n,:] + sum_{e: dst[e]==n} x[src[e],:] ),  F = 128.
// Pipeline: (1) out = x  (self-loop + init), (2) scatter-add via hardware
// no-return global_atomic_add_f32 at device scope, (3) relu in place.
// x (51.2MB) and out (51.2MB) are both L2-resident on MI455X (192MB L2),
// so phase (2) is bounded by L2 atomic-dword throughput; phases (1)/(3) are
// pure HBM bandwidth (~4us each at 23.3 TB/s).

#define F        128
#define TILE_E   256   // edges per block-tile
#define TPB      256   // 8 wave32s; one wave processes 32 edges of the tile

// Non-returning f32 atomic add, guaranteed to lower to the CDNA5 hardware
// instruction (STOREcnt path, no value returned -> no wait-for-return).
__device__ __forceinline__ void atom_add_f32(float* p, float v) {
    asm volatile("global_atomic_add_f32 %0, %1, off scope:SCOPE_DEV"
                 :: "v"(p), "v"(v) : "memory");
}

// ---------------------------------------------------------------- init: out = x
__global__ void gcn_init_kernel(const float* __restrict__ x,
                                float* __restrict__ out, int n4) {
    int i = blockIdx.x * blockDim.x + threadIdx.x;
    if (i < n4) {
        reinterpret_cast<float4*>(out)[i] =
            reinterpret_cast<const float4*>(x)[i];
    }
}

// ------------------------------------------------------- scatter: out[dst]+=x[src]
__global__ void gcn_scatter_kernel(const float* __restrict__ x,
                                   const int* __restrict__ esrc,
                                   const int* __restrict__ edst,
                                   float* __restrict__ out, int E) {
    __shared__ int s_src[TILE_E];
    __shared__ int s_dst[TILE_E];

    const int t    = threadIdx.x;
    const int lane = t & 31;
    const int wv   = t >> 5;
    const int base = blockIdx.x * TILE_E;
    const int n    = min(TILE_E, E - base);   // >= 1 by grid sizing
    const int tc   = min(t, n - 1);           // clamp tail so EXEC stays all-on

    // Stage this tile's edge indices into LDS with the CDNA5 async-copy path.
    // 256 lanes x b32 each for src and dst (coalesces to 128B/wave requests).
    {
        unsigned ls = (unsigned)(unsigned long long)(&s_src[t]);
        unsigned ld = (unsigned)(unsigned long long)(&s_dst[t]);
        const int* gs = esrc + base + tc;
        const int* gd = edst + base + tc;
        asm volatile("global_load_async_to_lds_b32 %0, %1, off"
                     :: "v"(ls), "v"(gs) : "memory");
        asm volatile("global_load_async_to_lds_b32 %0, %1, off"
                     :: "v"(ld), "v"(gd) : "memory");
        asm volatile("s_wait_asynccnt 0" ::: "memory");
    }
    __syncthreads();   // each wave reads back only the indices its own lanes staged

    // Wave wv owns edges [wv*32, wv*32+32) of the tile; lane owns a float4
    // (16B) of the 512B feature row -> one global_load_b128 + four no-return
    // global_atomic_add_f32 per lane per edge.  Software-pipelined: edge i+1's
    // source-row load is issued while edge i's atomics drain.
    const int lo = wv * 32;
    const int hi = min(n, lo + 32);
    if (lo < hi) {
        int    d = s_dst[lo];
        float4 v = *reinterpret_cast<const float4*>(
                       x + (size_t)s_src[lo] * F + lane * 4);
        for (int i = lo; i < hi; ++i) {
            const bool more = (i + 1 < hi);
            int    dn = d;
            float4 vn = v;
            if (more) {
                dn = s_dst[i + 1];
                vn = *reinterpret_cast<const float4*>(
                         x + (size_t)s_src[i + 1] * F + lane * 4);
                if (i + 2 < hi) {
                    // pull the row after next toward L2/L0 (global_prefetch_b8)
                    __builtin_prefetch(
                        x + (size_t)s_src[i + 2] * F + lane * 4, 0, 3);
                }
            }
            float* o = out + (size_t)d * F + lane * 4;
            atom_add_f32(o + 0, v.x);
            atom_add_f32(o + 1, v.y);
            atom_add_f32(o + 2, v.z);
            atom_add_f32(o + 3, v.w);
            d = dn;
            v = vn;
        }
    }
}

// ---------------------------------------------------------------- relu in place
__global__ void gcn_relu_kernel(float* __restrict__ out, int n4) {
    int i = blockIdx.x * blockDim.x + threadIdx.x;
    if (i < n4) {
        float4 v = reinterpret_cast<float4*>(out)[i];
        v.x = fmaxf(v.x, 0.0f);
        v.y = fmaxf(v.y, 0.0f);
        v.z = fmaxf(v.z, 0.0f);
        v.w = fmaxf(v.w, 0.0f);
        reinterpret_cast<float4*>(out)[i] = v;
    }
}

extern "C" void kernel_launch(void* const* d_in, const int* in_sizes, int n_in,
                              void* d_out, int out_size, void* d_ws, size_t ws_size,
                              hipStream_t stream) {
    (void)n_in; (void)d_ws; (void)ws_size;

    const float* x   = (const float*)d_in[0];          // [N, 128] f32
    const int*   ei  = (const int*)d_in[1];            // [2, E] int32 (flattened)
    float*       out = (float*)d_out;                  // [N, 128] f32

    const int NF = in_sizes[0];                        // N * 128
    const int E  = in_sizes[1] / 2;
    const int n4 = NF / 4;                             // float4 count

    const int* esrc = ei;          // edge_index[0]
    const int* edst = ei + E;      // edge_index[1]

    // (1) out = x  (covers the appended self-loops)
    gcn_init_kernel<<<(n4 + TPB - 1) / TPB, TPB, 0, stream>>>(x, out, n4);

    // (2) scatter-add x[src] into out[dst]
    const int tiles = (E + TILE_E - 1) / TILE_E;
    gcn_scatter_kernel<<<tiles, TPB, 0, stream>>>(x, esrc, edst, out, E);

    // (3) relu
    gcn_relu_kernel<<<(n4 + TPB - 1) / TPB, TPB, 0, stream>>>(out, n4);
}